// GCN2Layer_11081015623741
// MI455X (gfx1250) — compile-verified
//
#include <hip/hip_runtime.h>
#include <hip/hip_bf16.h>

// ---------------------------------------------------------------------------
// GCN 2-layer for MI455X (gfx1250, wave32, WMMA).
#define N_NODES 100000
#define N_EDGES 1600000
#define IN_CH   128
#define HID_CH  64
#define OUT_CH  32

typedef float v2f __attribute__((ext_vector_type(2)));
typedef float v8f __attribute__((ext_vector_type(8)));

// ---------------------------------------------------------------------------
// Utility: zero a float buffer (vectorized; all our sizes are /4).
__global__ __launch_bounds__(256) void zero_f32x4(float4* __restrict__ p, long long n4) {
    long long i = (long long)blockIdx.x * blockDim.x + threadIdx.x;
    if (i < n4) p[i] = make_float4(0.f, 0.f, 0.f, 0.f);
}

// ---------------------------------------------------------------------------
// Degree count via f32 atomics on dst.
__global__ __launch_bounds__(256) void degree_kernel(const int* __restrict__ dst,
                                                     float* __restrict__ deg, int E) {
    int e = blockIdx.x * blockDim.x + threadIdx.x;
    if (e < E) atomicAdd(&deg[dst[e]], 1.0f);
}

// dinv[i] = rsqrt(deg[i] + 1)  (self-loop); deg+1 >= 1 so no zero guard needed.
__global__ __launch_bounds__(256) void dinv_kernel(float* __restrict__ deg_inout, int N) {
    int i = blockIdx.x * blockDim.x + threadIdx.x;
    if (i < N) deg_inout[i] = __frsqrt_rn(deg_inout[i] + 1.0f);
}

// ---------------------------------------------------------------------------
// GEMM1: h1[N,64] = x[N,128] @ W1[128,64], pure f32 via V_WMMA_F32_16X16X4_F32.
// Block = 128 threads = 4 waves. Wave w computes the 16x16 tile at
// rows [blockIdx*16,+16), cols [w*16,+16). K loop: 128/4 = 32 WMMA ops.
//
// W is staged TRANSPOSED in LDS with padded row stride (K+4) so that
//  - the B operand (two adjacent K values for one column) is a single
//    8-byte-aligned ds_load_b64 (no register shuffles), and
//  - stride mod 64 == 4, so the 32 lanes of a wave hit all 64 LDS banks
//    conflict-free (lanes 0-15 banks =={0,1} mod 4, lanes 16-31 =={2,3}).
#define WT1 (IN_CH + 4)
__global__ __launch_bounds__(128) void gemm1_wmma(const float* __restrict__ x,
                                                  const float* __restrict__ W,
                                                  float* __restrict__ h) {
    __shared__ float Wt[HID_CH * WT1];     // 33 KB of the 320 KB/WGP LDS
    for (int i = threadIdx.x; i < IN_CH * HID_CH; i += 128) {
        int k = i / HID_CH, n = i % HID_CH;   // W row-major [128,64]
        Wt[n * WT1 + k] = W[i];
    }
    __syncthreads();

    const int wave = threadIdx.x >> 5;
    const int lane = threadIdx.x & 31;
    const int rowBase = blockIdx.x * 16;
    const int colBase = wave * 16;
    const int m    = lane & 15;            // A row / B col within tile
    const int koff = (lane >> 4) * 2;      // lanes 0-15: K=0,1 ; lanes 16-31: K=2,3

    const float* arow = x  + (size_t)(rowBase + m) * IN_CH + koff;
    const float* brow = Wt + (colBase + m) * WT1 + koff;
    v8f c = {};
    #pragma unroll
    for (int k = 0; k < IN_CH; k += 4) {
        v2f a = *(const v2f*)(arow + k);   // global_load_b64
        v2f b = *(const v2f*)(brow + k);   // ds_load_b64, bank-conflict-free
        c = __builtin_amdgcn_wmma_f32_16x16x4_f32(
                /*neg_a=*/false, a, /*neg_b=*/false, b,
                /*c_mod=*/(short)0, c, /*reuse_a=*/false, /*reuse_b=*/false);
    }
    // C/D layout: VGPR v -> row v (lanes 0-15) or v+8 (lanes 16-31), col = lane&15.
    const int rsel = (lane >> 4) * 8;
    #pragma unroll
    for (int v = 0; v < 8; ++v)
        h[(size_t)(rowBase + rsel + v) * HID_CH + colBase + m] = c[v];
}

// ---------------------------------------------------------------------------
// GEMM2: h2[N,32] = z1[N,64] @ W2[64,32]. Block = 128 threads = 4 waves.
// Wave w: rows rowBase + (w>>1)*16, cols (w&1)*16. K loop: 64/4 = 16 WMMAs.
#define WT2 (HID_CH + 4)
__global__ __launch_bounds__(128) void gemm2_wmma(const float* __restrict__ z,
                                                  const float* __restrict__ W,
                                                  float* __restrict__ h) {
    __shared__ float Wt[OUT_CH * WT2];     // 8.7 KB
    for (int i = threadIdx.x; i < HID_CH * OUT_CH; i += 128) {
        int k = i / OUT_CH, n = i % OUT_CH;   // W row-major [64,32]
        Wt[n * WT2 + k] = W[i];
    }
    __syncthreads();

    const int wave = threadIdx.x >> 5;
    const int lane = threadIdx.x & 31;
    const int rowBase = blockIdx.x * 32 + (wave >> 1) * 16;
    const int colBase = (wave & 1) * 16;
    const int m    = lane & 15;
    const int koff = (lane >> 4) * 2;

    const float* arow = z  + (size_t)(rowBase + m) * HID_CH + koff;
    const float* brow = Wt + (colBase + m) * WT2 + koff;
    v8f c = {};
    #pragma unroll
    for (int k = 0; k < HID_CH; k += 4) {
        v2f a = *(const v2f*)(arow + k);
        v2f b = *(const v2f*)(brow + k);
        c = __builtin_amdgcn_wmma_f32_16x16x4_f32(
                false, a, false, b, (short)0, c, false, false);
    }
    const int rsel = (lane >> 4) * 8;
    #pragma unroll
    for (int v = 0; v < 8; ++v)
        h[(size_t)(rowBase + rsel + v) * OUT_CH + colBase + m] = c[v];
}

// ---------------------------------------------------------------------------
// Edge scatter, layer 1: 16 threads per edge, float4 (128-bit) gather from h1
// (L2-resident: 25.6 MB << 192 MB L2) + 4 independent global f32 atomics.
__global__ __launch_bounds__(256) void scatter64(const int* __restrict__ src,
                                                 const int* __restrict__ dst,
                                                 const float* __restrict__ dinv,
                                                 const float* __restrict__ h,
                                                 float* __restrict__ agg, int E) {
    long long t = (long long)blockIdx.x * blockDim.x + threadIdx.x;
    int e = (int)(t >> 4);
    if (e >= E) return;
    int c = ((int)t & 15) * 4;
    int s = src[e], d = dst[e];
    float norm = dinv[s] * dinv[d];
    float4 v = *(const float4*)(h + (size_t)s * HID_CH + c);   // global_load_b128
    float* ap = agg + (size_t)d * HID_CH + c;
    atomicAdd(ap + 0, v.x * norm);
    atomicAdd(ap + 1, v.y * norm);
    atomicAdd(ap + 2, v.z * norm);
    atomicAdd(ap + 3, v.w * norm);
}

// Edge scatter, layer 2: 8 threads per edge over 32 features.
__global__ __launch_bounds__(256) void scatter32(const int* __restrict__ src,
                                                 const int* __restrict__ dst,
                                                 const float* __restrict__ dinv,
                                                 const float* __restrict__ h,
                                                 float* __restrict__ agg, int E) {
    long long t = (long long)blockIdx.x * blockDim.x + threadIdx.x;
    int e = (int)(t >> 3);
    if (e >= E) return;
    int c = ((int)t & 7) * 4;
    int s = src[e], d = dst[e];
    float norm = dinv[s] * dinv[d];
    float4 v = *(const float4*)(h + (size_t)s * OUT_CH + c);
    float* ap = agg + (size_t)d * OUT_CH + c;
    atomicAdd(ap + 0, v.x * norm);
    atomicAdd(ap + 1, v.y * norm);
    atomicAdd(ap + 2, v.z * norm);
    atomicAdd(ap + 3, v.w * norm);
}

// ---------------------------------------------------------------------------
// Finalize: out = relu(agg + h*dinv^2 + bias). In-place on agg. F = 64 or 32.
template <int F>
__global__ __launch_bounds__(256) void finalize_kernel(const float* __restrict__ h,
                                                       const float* __restrict__ dinv,
                                                       const float* __restrict__ bias,
                                                       float* __restrict__ agg_inout,
                                                       int N) {
    long long t = (long long)blockIdx.x * blockDim.x + threadIdx.x;
    int node = (int)(t / (F / 4));
    if (node >= N) return;
    int c = ((int)t % (F / 4)) * 4;
    float di = dinv[node];
    float sl = di * di;
    float4 a  = *(const float4*)(agg_inout + (size_t)node * F + c);
    float4 hv = *(const float4*)(h         + (size_t)node * F + c);
    float4 b  = *(const float4*)(bias + c);
    float4 o;
    o.x = fmaxf(a.x + hv.x * sl + b.x, 0.f);
    o.y = fmaxf(a.y + hv.y * sl + b.y, 0.f);
    o.z = fmaxf(a.z + hv.z * sl + b.z, 0.f);
    o.w = fmaxf(a.w + hv.w * sl + b.w, 0.f);
    *(float4*)(agg_inout + (size_t)node * F + c) = o;
}

// ---------------------------------------------------------------------------
extern "C" void kernel_launch(void* const* d_in, const int* in_sizes, int n_in,
                              void* d_out, int out_size, void* d_ws, size_t ws_size,
                              hipStream_t stream) {
    const float* x  = (const float*)d_in[0];
    const int*   ei = (const int*)d_in[1];     // [2, E] row-major
    const float* W1 = (const float*)d_in[2];
    const float* b1 = (const float*)d_in[3];
    const float* W2 = (const float*)d_in[4];
    const float* b2 = (const float*)d_in[5];
    float* out = (float*)d_out;

    const int N = N_NODES, E = N_EDGES;
    const int* src = ei;
    const int* dst = ei + E;

    // Workspace layout (bytes), 512-aligned.
    char* ws = (char*)d_ws;
    size_t off = 0;
    auto alloc = [&](size_t bytes) { char* p = ws + off; off = (off + bytes + 511) & ~(size_t)511; return p; };
    float* dinv = (float*)alloc((size_t)N * 4);             // deg -> dinv in place
    float* h1   = (float*)alloc((size_t)N * HID_CH * 4);    // 25.6 MB
    float* z1   = (float*)alloc((size_t)N * HID_CH * 4);    // agg1 -> z1 in place
    float* h2   = (float*)alloc((size_t)N * OUT_CH * 4);    // 12.8 MB
    (void)ws_size;

    // 1) zero accumulators (ws/d_out are poisoned, zero every call).
    {
        long long n4 = N / 4;
        zero_f32x4<<<(unsigned)((n4 + 255) / 256), 256, 0, stream>>>((float4*)dinv, n4);
        n4 = (long long)N * HID_CH / 4;
        zero_f32x4<<<(unsigned)((n4 + 255) / 256), 256, 0, stream>>>((float4*)z1, n4);
        n4 = (long long)N * OUT_CH / 4;
        zero_f32x4<<<(unsigned)((n4 + 255) / 256), 256, 0, stream>>>((float4*)out, n4);
    }

    // 2) degree + dinv
    degree_kernel<<<(E + 255) / 256, 256, 0, stream>>>(dst, dinv, E);
    dinv_kernel<<<(N + 255) / 256, 256, 0, stream>>>(dinv, N);

    // 3) layer 1: WMMA GEMM -> scatter -> finalize(relu)
    gemm1_wmma<<<N / 16, 128, 0, stream>>>(x, W1, h1);
    {
        long long nt = (long long)E * 16;
        scatter64<<<(unsigned)((nt + 255) / 256), 256, 0, stream>>>(src, dst, dinv, h1, z1, E);
        nt = (long long)N * (HID_CH / 4);
        finalize_kernel<HID_CH><<<(unsigned)((nt + 255) / 256), 256, 0, stream>>>(h1, dinv, b1, z1, N);
    }

    // 4) layer 2: WMMA GEMM -> scatter (into d_out) -> finalize(relu)
    gemm2_wmma<<<N / 32, 128, 0, stream>>>(z1, W2, h2);
    {
        long long nt = (long long)E * 8;
        scatter32<<<(unsigned)((nt + 255) / 256), 256, 0, stream>>>(src, dst, dinv, h2, out, E);
        nt = (long long)N * (OUT_CH / 4);
        finalize_kernel<OUT_CH><<<(unsigned)((nt + 255) / 256), 256, 0, stream>>>(h2, dinv, b2, out, N);
    }
}